// VCTreePredictor_28441273434456
// MI455X (gfx1250) — compile-verified
//
#include <hip/hip_runtime.h>

// Problem constants (from the reference)
#define N_OBJ     8192
#define N_REL     32768
#define HID       512
#define POOL      4096
#define N_OBJ_CLS 151
#define N_REL_CLS 51
#define NPAD      64      // N_REL_CLS padded to 4 WMMA n-tiles

typedef float v2f __attribute__((ext_vector_type(2)));
typedef float v8f __attribute__((ext_vector_type(8)));

// ---------------------------------------------------------------------------
// gfx1250 async Global->LDS copy (ASYNCcnt-tracked, bypasses VGPRs).
// The clang builtins type their pointer args in the __device__ LangAS which
// HIP source cannot spell, so we emit the instruction via inline asm:
//   global_load_async_to_lds_b128 vdst(LDS byte addr), vaddr(64b global), off
// LDS byte address = low 32 bits of the flat pointer (ISA: LDS aperture maps
// the low 32 address bits, wave-relative).
// ---------------------------------------------------------------------------
#if defined(__gfx1250__)
#define USE_ASYNC 1
__device__ __forceinline__ void async_copy16(const float* g, float* lds) {
  unsigned l = (unsigned)(unsigned long long)lds;
  asm volatile("global_load_async_to_lds_b128 %0, %1, off"
               :
               : "v"(l), "v"(g)
               : "memory");
}
__device__ __forceinline__ void async_wait0() {
#if __has_builtin(__builtin_amdgcn_s_wait_asynccnt)
  __builtin_amdgcn_s_wait_asynccnt(0);
#else
  asm volatile("s_wait_asynccnt 0" ::: "memory");
#endif
}
#else
#define USE_ASYNC 0
#endif

// ---------------------------------------------------------------------------
// Transpose: dst[n][k] = src[k][n]   (src is [K][N]); coalesced on dst.
// Used once on W_post_emb (512x1024) and W_post_cat (1024x4096) so GEMM
// B-fragments become contiguous k-major 8B loads.
// ---------------------------------------------------------------------------
__global__ __launch_bounds__(256) void transpose_f32(const float* __restrict__ src,
                                                     float* __restrict__ dst,
                                                     int K, int N) {
  size_t t = (size_t)blockIdx.x * 256 + threadIdx.x;  // 0 .. N*K-1
  int n = (int)(t / K);
  int k = (int)(t % K);
  dst[t] = src[(size_t)k * N + n];
}

// ---------------------------------------------------------------------------
// W_ctx [4096,51] -> transposed + zero-padded [64][4096] (k-major, 1 MB,
// L2-resident). Fused kernel B-fragments become single 8B loads.
// ---------------------------------------------------------------------------
__global__ __launch_bounds__(256) void pad_wctx_t_kernel(const float* __restrict__ w,
                                                         float* __restrict__ wpt) {
  int t = blockIdx.x * 256 + threadIdx.x;   // t = c*4096 + r
  int c = t >> 12;                          // 0..63  (padded class)
  int r = t & 4095;                         // 0..4095 (k)
  wpt[t] = (c < N_REL_CLS) ? w[r * N_REL_CLS + c] : 0.0f;
}

// ---------------------------------------------------------------------------
// fp32 WMMA GEMM: C[M,N] = act(A[M,K] @ B[K,N] + bias), with B supplied
// TRANSPOSED (Bt[n][k], leading dim ldbt). Block tile 128(M) x 64(N),
// 256 threads = 8 waves (4x2), each wave 32x32 via 4x v_wmma_f32_16x16x4_f32.
// K staged through LDS in slices of 16 using async Global->LDS when available.
// ---------------------------------------------------------------------------
__global__ __launch_bounds__(256) void gemm_f32_wmma(
    const float* __restrict__ A,  int lda,
    const float* __restrict__ Bt, int ldbt,
    float* __restrict__ C, int ldc,
    int K, const float* __restrict__ bias, int relu)
{
  __shared__ float As[128 * 16];  // [m][k], 8 KB
  __shared__ float Bs[64 * 16];   // [n][k] (k-major!), 4 KB

  const int tid  = threadIdx.x;
  const int wave = tid >> 5;
  const int lane = tid & 31;
  const int lo   = lane & 15;
  const int hi   = lane >> 4;
  const int wm   = wave >> 1;   // 0..3 -> m sub-block
  const int wn   = wave & 1;    // 0..1 -> n sub-block
  const size_t row0 = (size_t)blockIdx.y * 128;
  const size_t col0 = (size_t)blockIdx.x * 64;

  v8f acc[2][2] = {{{0.f,0.f,0.f,0.f,0.f,0.f,0.f,0.f},
                    {0.f,0.f,0.f,0.f,0.f,0.f,0.f,0.f}},
                   {{0.f,0.f,0.f,0.f,0.f,0.f,0.f,0.f},
                    {0.f,0.f,0.f,0.f,0.f,0.f,0.f,0.f}}};

  for (int k0 = 0; k0 < K; k0 += 16) {
#if USE_ASYNC
    // A: 128 rows x 16 floats = 512 x 16B chunks -> 2 per thread
    #pragma unroll
    for (int i = 0; i < 2; ++i) {
      int t  = tid + i * 256;
      int m  = t >> 2;
      int kq = (t & 3) * 4;
      async_copy16(&A[(row0 + m) * (size_t)lda + k0 + kq], &As[m * 16 + kq]);
    }
    // Bt: 64 rows x 16 floats = 256 x 16B chunks -> 1 per thread
    {
      int n  = tid >> 2;
      int kq = (tid & 3) * 4;
      async_copy16(&Bt[(col0 + n) * (size_t)ldbt + k0 + kq], &Bs[n * 16 + kq]);
    }
    async_wait0();
#else
    #pragma unroll
    for (int i = 0; i < 2; ++i) {
      int t  = tid + i * 256;
      int m  = t >> 2;
      int kq = (t & 3) * 4;
      *(float4*)&As[m * 16 + kq] =
          *(const float4*)&A[(row0 + m) * (size_t)lda + k0 + kq];
    }
    {
      int n  = tid >> 2;
      int kq = (tid & 3) * 4;
      *(float4*)&Bs[n * 16 + kq] =
          *(const float4*)&Bt[(col0 + n) * (size_t)ldbt + k0 + kq];
    }
#endif
    __syncthreads();

    #pragma unroll
    for (int kk = 0; kk < 16; kk += 4) {
      // A fragment: a[v] = A[m][kk + v + 2*hi]  -> contiguous 8B LDS load
      v2f a[2];
      #pragma unroll
      for (int i = 0; i < 2; ++i) {
        int m = wm * 32 + i * 16 + lo;
        a[i] = *(const v2f*)&As[m * 16 + kk + 2 * hi];
      }
      #pragma unroll
      for (int j = 0; j < 2; ++j) {
        // B fragment: b[v] = B[kk + v + 2*hi][n] -> contiguous in k-major Bs
        int n = wn * 32 + j * 16 + lo;
        v2f b = *(const v2f*)&Bs[n * 16 + kk + 2 * hi];
        #pragma unroll
        for (int i = 0; i < 2; ++i) {
          acc[i][j] = __builtin_amdgcn_wmma_f32_16x16x4_f32(
              false, a[i], false, b, (short)0, acc[i][j], false, false);
        }
      }
    }
    __syncthreads();
  }

  // Store: C element (m = v + 8*hi, n = lo) per accumulator
  #pragma unroll
  for (int i = 0; i < 2; ++i) {
    #pragma unroll
    for (int j = 0; j < 2; ++j) {
      size_t m0 = row0 + wm * 32 + i * 16;
      size_t n  = col0 + wn * 32 + j * 16 + lo;
      float bv = bias ? bias[n] : 0.0f;
      #pragma unroll
      for (int v = 0; v < 8; ++v) {
        size_t m = m0 + v + 8 * hi;
        float x = acc[i][j][v] + bv;
        if (relu) x = fmaxf(x, 0.0f);
        C[m * (size_t)ldc + n] = x;
      }
    }
  }
}

// ---------------------------------------------------------------------------
// Fused per-relation kernel. Block = 16 relations, 256 threads = 8 waves.
// For each 1024-wide chunk of POOL:
//   stage PU[16][1024] = (Hp[i0] + Tp[i1] + b_post_cat) * union  into LDS
//   wave w contracts k-slice [w*128, w*128+128) against k-major padded W_ctx
//   with v_wmma_f32_16x16x4_f32 into a private [16 x 64] accumulator.
// Cross-wave LDS reduction, + b_ctx + freq_table[obj pair], store 16x51.
// Dynamic LDS: 16*1024 floats = 64 KB.
// ---------------------------------------------------------------------------
__global__ __launch_bounds__(256) void fused_rel_kernel(
    const float* __restrict__ hp,    const float* __restrict__ tp,
    const float* __restrict__ bpc,   const float* __restrict__ uf,
    const float* __restrict__ wpadt, const float* __restrict__ bctx,
    const float* __restrict__ freq,  const int*   __restrict__ rel,
    const int*   __restrict__ obj,   float* __restrict__ out)
{
  extern __shared__ float smem[];   // 16*1024 floats

  const int tid  = threadIdx.x;
  const int wave = tid >> 5;
  const int lane = tid & 31;
  const int lo   = lane & 15;
  const int hi   = lane >> 4;
  const size_t row0 = (size_t)blockIdx.x * 16;

  v8f acc[4] = {{0.f,0.f,0.f,0.f,0.f,0.f,0.f,0.f},
                {0.f,0.f,0.f,0.f,0.f,0.f,0.f,0.f},
                {0.f,0.f,0.f,0.f,0.f,0.f,0.f,0.f},
                {0.f,0.f,0.f,0.f,0.f,0.f,0.f,0.f}};

  for (int ch = 0; ch < POOL / 1024; ++ch) {
    const int cb = ch * 1024;
    __syncthreads();  // previous chunk fully consumed before overwrite
    // Stage PU chunk: one row per iteration, 256 threads cover 1024 cols
    #pragma unroll 1
    for (int m = 0; m < 16; ++m) {
      const int i0 = rel[(row0 + m) * 2 + 0];   // uniform -> scalar loads
      const int i1 = rel[(row0 + m) * 2 + 1];
      const int nc = cb + tid * 4;
      float4 h4 = *(const float4*)&hp[(size_t)i0 * POOL + nc];
      float4 t4 = *(const float4*)&tp[(size_t)i1 * POOL + nc];
      float4 b4 = *(const float4*)&bpc[nc];
      float4 u4 = *(const float4*)&uf[(row0 + m) * (size_t)POOL + nc];
      float4 r;
      r.x = (h4.x + t4.x + b4.x) * u4.x;
      r.y = (h4.y + t4.y + b4.y) * u4.y;
      r.z = (h4.z + t4.z + b4.z) * u4.z;
      r.w = (h4.w + t4.w + b4.w) * u4.w;
      *(float4*)&smem[m * 1024 + tid * 4] = r;
    }
    __syncthreads();

    const int kb = wave * 128;   // wave-private k-slice within the chunk
    #pragma unroll 4
    for (int kk = 0; kk < 128; kk += 4) {
      // A fragment from LDS: a[v] = PU[lo][kb + kk + v + 2*hi]
      v2f a = *(const v2f*)&smem[lo * 1024 + kb + kk + 2 * hi];
      const size_t kg = (size_t)(cb + kb + kk + 2 * hi);
      #pragma unroll
      for (int j = 0; j < 4; ++j) {
        int n = j * 16 + lo;
        // B fragment: single contiguous 8B load from k-major padded W_ctx
        v2f b = *(const v2f*)&wpadt[(size_t)n * POOL + kg];
        acc[j] = __builtin_amdgcn_wmma_f32_16x16x4_f32(
            false, a, false, b, (short)0, acc[j], false, false);
      }
    }
  }
  __syncthreads();

  // Write per-wave partials [16 x 64] into LDS (8*1024 floats, reusing smem)
  #pragma unroll
  for (int j = 0; j < 4; ++j) {
    int n = j * 16 + lo;
    #pragma unroll
    for (int v = 0; v < 8; ++v) {
      int m = v + 8 * hi;
      smem[(wave * 16 + m) * 64 + n] = acc[j][v];
    }
  }
  __syncthreads();

  // Reduce 8 partials, add class bias + frequency prior, store 16 x 51
  #pragma unroll
  for (int i = 0; i < 4; ++i) {
    int t = tid + i * 256;    // 0..1023 covers [16][64]
    int m = t >> 6;
    int c = t & 63;
    float s = 0.0f;
    #pragma unroll
    for (int w = 0; w < 8; ++w) s += smem[(w * 16 + m) * 64 + c];
    if (c < N_REL_CLS) {
      int i0 = rel[(row0 + m) * 2 + 0];
      int i1 = rel[(row0 + m) * 2 + 1];
      int p  = obj[i0] * N_OBJ_CLS + obj[i1];
      out[(row0 + m) * N_REL_CLS + c] = s + bctx[c] + freq[(size_t)p * N_REL_CLS + c];
    }
  }
}

// ---------------------------------------------------------------------------
// Launch. Workspace layout (floats):
//   er    @ 0         : 8192*1024  edge_rep (relu'd)            32 MB
//   hp    @ 8388608   : 8192*4096  head_rep @ Wpc[0:512]       128 MB
//   tp    @ 41943040  : 8192*4096  tail_rep @ Wpc[512:1024]    128 MB
//   wpadt @ 75497472  : 64*4096    W_ctx^T zero-padded           1 MB
//   wpet  @ 75759616  : 1024*512   W_post_emb^T                  2 MB
//   wpct  @ 76283904  : 4096*1024  W_post_cat^T                 16 MB
// Total ~307 MB (inputs alone are >530 MB; scratch is proportionate).
// ---------------------------------------------------------------------------
extern "C" void kernel_launch(void* const* d_in, const int* in_sizes, int n_in,
                              void* d_out, int out_size, void* d_ws, size_t ws_size,
                              hipStream_t stream) {
  const float* edge_ctx = (const float*)d_in[0];
  const float* uf       = (const float*)d_in[1];
  const int*   rel      = (const int*)d_in[2];
  const int*   obj      = (const int*)d_in[3];
  const float* Wpe      = (const float*)d_in[4];
  const float* bpe      = (const float*)d_in[5];
  const float* Wpc      = (const float*)d_in[6];
  const float* bpc      = (const float*)d_in[7];
  const float* Wctx     = (const float*)d_in[8];
  const float* bctx     = (const float*)d_in[9];
  const float* freq     = (const float*)d_in[10];
  float* out = (float*)d_out;

  float* ws    = (float*)d_ws;
  float* er    = ws;
  float* hp    = ws + (size_t)8192 * 1024;
  float* tp    = hp + (size_t)8192 * 4096;
  float* wpadt = tp + (size_t)8192 * 4096;
  float* wpet  = wpadt + (size_t)NPAD * POOL;
  float* wpct  = wpet + (size_t)1024 * 512;

  // K0: one-time weight reshapes (k-major B for all WMMA consumers)
  pad_wctx_t_kernel<<<dim3((POOL * NPAD) / 256), dim3(256), 0, stream>>>(Wctx, wpadt);
  transpose_f32<<<dim3((1024 * 512) / 256), dim3(256), 0, stream>>>(Wpe, wpet, HID, 2 * HID);
  transpose_f32<<<dim3((4096 * 1024) / 256), dim3(256), 0, stream>>>(Wpc, wpct, 2 * HID, POOL);

  // K1: edge_rep = relu(edge_ctx @ W_post_emb + b)   [8192 x 1024]
  gemm_f32_wmma<<<dim3(1024 / 64, 8192 / 128), dim3(256), 0, stream>>>(
      edge_ctx, HID, wpet, HID, er, 2 * HID, HID, bpe, 1);

  // K2a: Hp = head_rep @ Wpc[0:512, :]    [8192 x 4096]
  gemm_f32_wmma<<<dim3(POOL / 64, 8192 / 128), dim3(256), 0, stream>>>(
      er, 2 * HID, wpct, 2 * HID, hp, POOL, HID, nullptr, 0);

  // K2b: Tp = tail_rep @ Wpc[512:1024, :] [8192 x 4096]
  gemm_f32_wmma<<<dim3(POOL / 64, 8192 / 128), dim3(256), 0, stream>>>(
      er + HID, 2 * HID, wpct + HID, 2 * HID, tp, POOL, HID, nullptr, 0);

  // K3: fused gather + add-bias + elementwise-union + GEMM(4096->51) + freq
  fused_rel_kernel<<<dim3(N_REL / 16), dim3(256), 16 * 1024 * sizeof(float), stream>>>(
      hp, tp, bpc, uf, wpadt, bctx, freq, rel, obj, out);
}